// GraphAttentionBlock_57913339019510
// MI455X (gfx1250) — compile-verified
//
#include <hip/hip_runtime.h>
#include <hip/hip_bf16.h>
#include <math.h>

// ---------------------------------------------------------------------------
// Fused graph-attention block for MI455X (gfx1250, wave32).
// v2: 4 frames per block, joints densely packed into 68 rows (5 M-tiles);
// each wave loads one B fragment per k-step and reuses it across all 5
// M-tiles (register-level B reuse) -> ~8x less L2 weight traffic than v1.
// All GEMMs run on V_WMMA_F32_16X16X4_F32 (fp32, matches reference numerics).
// ---------------------------------------------------------------------------

typedef __attribute__((ext_vector_type(2))) float v2f;
typedef __attribute__((ext_vector_type(8))) float v8f;

#define N_J     17
#define D_IN    128
#define T_LEN   512
#define B_SZ    32
#define H_HEADS 16
#define HD      8
#define C2      256            // 2*D
#define NEG_INF_F (-9e15f)
#define BN_EPS  1e-5f

#define FRAMES  4              // t-frames per block
#define ROWS_R  (FRAMES * N_J) // 68 real rows
#define M_TILES 5              // ceil(68/16) -> 80 padded rows
#define ROWS_P  (M_TILES * 16)

#define CATA_STRIDE 260        // [row][0..255]: 0-127 xp | 128-255 attn out
#define QKV_STRIDE  388        // [row][0..383]: q | k | v (h*8+e within each)

__global__ __launch_bounds__(256) void gab_fused_kernel(
    const float* __restrict__ x,        // (B, D, T, N)
    const float* __restrict__ adj,      // (N, N)
    const float* __restrict__ Wq,       // (H, D, hd)
    const float* __restrict__ Wk,       // (H, D, hd)
    const float* __restrict__ Wv,       // (H, D, hd)
    const float* __restrict__ adj2,     // (H, N, N)
    const float* __restrict__ conv_w,   // (2D, 2D)
    const float* __restrict__ bn_gamma,
    const float* __restrict__ bn_beta,
    const float* __restrict__ bn_mean,
    const float* __restrict__ bn_var,
    float* __restrict__ out)            // (B, 2D, T, N)
{
  __shared__ float catA[ROWS_P * CATA_STRIDE]; // 83.2 KB
  __shared__ float qkv[ROWS_R * QKV_STRIDE];   // 105.5 KB

  const int t0   = blockIdx.x * FRAMES;
  const int b    = blockIdx.y;
  const int tid  = threadIdx.x;
  const int lane = tid & 31;
  const int wave = tid >> 5;

  const int laneRow = lane & 15;           // M (A) / N (B,C/D) index in tile
  const int kPhase  = (lane < 16) ? 0 : 2; // K phase split across lane halves

  // ---- zero pad rows 68..79 of catA ----
  for (int i = tid; i < (ROWS_P - ROWS_R) * CATA_STRIDE; i += 256)
    catA[ROWS_R * CATA_STRIDE + i] = 0.0f;

  // ---- load 4 frames of x, transposing (d, n) -> catA[f*17+n][d] ----
  for (int f = 0; f < FRAMES; ++f) {
    const long xbase = (long)b * (D_IN * T_LEN * N_J) + (long)(t0 + f) * N_J;
    for (int i = tid; i < N_J * D_IN; i += 256) {
      int d = i / N_J;
      int n = i - d * N_J;
      catA[(f * N_J + n) * CATA_STRIDE + d] = x[xbase + (long)d * (T_LEN * N_J) + n];
    }
  }
  __syncthreads();

  // ------------------------- QKV projection GEMM ---------------------------
  // cols 0..383 (24 tiles of 16), K = 128. One B fragment per k-step reused
  // over all 5 M-tiles.
  for (int nt = wave; nt < 24; nt += 8) {
    const int mycol = nt * 16 + laneRow;
    // per-lane weight base: W[h*D*HD + e], element for k at wbase[k*HD]
    const int seg = mycol >> 7;             // 0=q 1=k 2=v
    const int c   = mycol & 127;
    const float* wsel = (seg == 0) ? Wq : ((seg == 1) ? Wk : Wv);
    const float* wbase = wsel + (c >> 3) * (D_IN * HD) + (c & 7);

    v8f acc[M_TILES];
#pragma unroll
    for (int mt = 0; mt < M_TILES; ++mt) acc[mt] = (v8f){};

#pragma unroll 4
    for (int k0 = 0; k0 < D_IN; k0 += 4) {
      v2f bb;
      bb.x = wbase[(k0 + kPhase) * HD];
      bb.y = wbase[(k0 + kPhase + 1) * HD];
#pragma unroll
      for (int mt = 0; mt < M_TILES; ++mt) {
        const float* arow = &catA[(mt * 16 + laneRow) * CATA_STRIDE + k0 + kPhase];
        v2f a; a.x = arow[0]; a.y = arow[1];
        acc[mt] = __builtin_amdgcn_wmma_f32_16x16x4_f32(false, a, false, bb,
                                                        (short)0, acc[mt],
                                                        false, false);
      }
    }
#pragma unroll
    for (int mt = 0; mt < M_TILES; ++mt) {
#pragma unroll
      for (int j = 0; j < 8; ++j) {
        int row = mt * 16 + ((lane < 16) ? j : 8 + j);
        if (mt < M_TILES - 1 || row < ROWS_R)   // tiles 0..3 unconditional
          qkv[row * QKV_STRIDE + mycol] = acc[mt][j];
      }
    }
  }
  __syncthreads();

  // --------------------- masked multi-head attention -----------------------
  const float inv_sqrt_hd = 0.35355339059327373f; // 1/sqrt(8)
  for (int idx = tid; idx < FRAMES * H_HEADS * N_J; idx += 256) {
    int f   = idx / (H_HEADS * N_J);
    int rem = idx - f * (H_HEADS * N_J);
    int h   = rem / N_J;
    int n   = rem - h * N_J;
    int rb  = f * N_J;                      // row base of this frame

    float q[HD];
#pragma unroll
    for (int e = 0; e < HD; ++e) q[e] = qkv[(rb + n) * QKV_STRIDE + h * HD + e];

    float ev[N_J];
    float mx = -INFINITY;
#pragma unroll
    for (int m = 0; m < N_J; ++m) {
      float dot = 0.0f;
#pragma unroll
      for (int e = 0; e < HD; ++e)
        dot += q[e] * qkv[(rb + m) * QKV_STRIDE + 128 + h * HD + e];
      // symmetrized adjacency: a = 0.5*((adj+adj2)+(adj+adj2)^T); mask = a>0
      float am = adj[n * N_J + m] + adj[m * N_J + n]
               + adj2[(h * N_J + n) * N_J + m]
               + adj2[(h * N_J + m) * N_J + n];
      float val = (am > 0.0f) ? dot * inv_sqrt_hd : NEG_INF_F;
      ev[m] = val;
      mx = fmaxf(mx, val);
    }
    float s = 0.0f;
#pragma unroll
    for (int m = 0; m < N_J; ++m) { ev[m] = __expf(ev[m] - mx); s += ev[m]; }
    float inv_s = 1.0f / s;
#pragma unroll
    for (int e = 0; e < HD; ++e) {
      float accv = 0.0f;
#pragma unroll
      for (int m = 0; m < N_J; ++m)
        accv += ev[m] * qkv[(rb + m) * QKV_STRIDE + 256 + h * HD + e];
      catA[(rb + n) * CATA_STRIDE + 128 + h * HD + e] = accv * inv_s;
    }
  }
  __syncthreads();

  // ---------------- concat -> conv1x1 -> BN(eval) -> ReLU -------------------
  // y[row,o] = sum_c catA[row][c] * conv_w[o*256+c]; 16 O-tiles, K = 256.
  for (int ot = wave; ot < 16; ot += 8) {
    const int o = ot * 16 + laneRow;
    const float* bcol = &conv_w[o * C2];

    v8f acc[M_TILES];
#pragma unroll
    for (int mt = 0; mt < M_TILES; ++mt) acc[mt] = (v8f){};

#pragma unroll 4
    for (int k0 = 0; k0 < C2; k0 += 4) {
      v2f bb;
      bb.x = bcol[k0 + kPhase];
      bb.y = bcol[k0 + kPhase + 1];
#pragma unroll
      for (int mt = 0; mt < M_TILES; ++mt) {
        const float* arow = &catA[(mt * 16 + laneRow) * CATA_STRIDE + k0 + kPhase];
        v2f a; a.x = arow[0]; a.y = arow[1];
        acc[mt] = __builtin_amdgcn_wmma_f32_16x16x4_f32(false, a, false, bb,
                                                        (short)0, acc[mt],
                                                        false, false);
      }
    }

    const float scale = rsqrtf(bn_var[o] + BN_EPS) * bn_gamma[o];
    const float mean  = bn_mean[o];
    const float beta  = bn_beta[o];
    const long obase  = ((long)b * C2 + o) * (T_LEN * N_J);

#pragma unroll
    for (int mt = 0; mt < M_TILES; ++mt) {
#pragma unroll
      for (int j = 0; j < 8; ++j) {
        int row = mt * 16 + ((lane < 16) ? j : 8 + j);
        if (mt < M_TILES - 1 || row < ROWS_R) {
          int f = row / N_J;
          int n = row - f * N_J;
          float y = (acc[mt][j] - mean) * scale + beta;
          out[obase + (long)(t0 + f) * N_J + n] = fmaxf(y, 0.0f);
        }
      }
    }
  }
}

extern "C" void kernel_launch(void* const* d_in, const int* in_sizes, int n_in,
                              void* d_out, int out_size, void* d_ws, size_t ws_size,
                              hipStream_t stream) {
  (void)in_sizes; (void)n_in; (void)d_ws; (void)ws_size; (void)out_size;
  dim3 grid(T_LEN / FRAMES, B_SZ);   // (128, 32)
  gab_fused_kernel<<<grid, 256, 0, stream>>>(
      (const float*)d_in[0],  // x
      (const float*)d_in[1],  // adj
      (const float*)d_in[2],  // Wq
      (const float*)d_in[3],  // Wk
      (const float*)d_in[4],  // Wv
      (const float*)d_in[5],  // adj2
      (const float*)d_in[6],  // conv_w
      (const float*)d_in[7],  // bn_gamma
      (const float*)d_in[8],  // bn_beta
      (const float*)d_in[9],  // bn_mean
      (const float*)d_in[10], // bn_var
      (float*)d_out);
}